// FuseLayer_60971355734151
// MI455X (gfx1250) — compile-verified
//
#include <hip/hip_runtime.h>
#include <math.h>

// CDNA5 / gfx1250: wave32, WMMA 16x16 shapes, async global->LDS DMA.
typedef __attribute__((ext_vector_type(2))) float v2f;
typedef __attribute__((ext_vector_type(8))) float v8f;
typedef int v4i __attribute__((vector_size(16)));   // matches builtin param type

#define B_DIM   64
#define IN_DIM  1024
#define OUT_DIM 1024
#define E_DIM   8
#define KT      32      // K-tile staged in LDS per iteration
#define LDSW    36      // padded row stride (floats): 144B, 16B-aligned, conflict-free

#define AS1 __attribute__((address_space(1)))
#define AS3 __attribute__((address_space(3)))

// ---- CDNA5 async global->LDS copy (GLOBAL_LOAD_ASYNC_TO_LDS_B128, ASYNCcnt) ----
__device__ __forceinline__ void async_copy_b128(const float* g, float* l) {
#if __has_builtin(__builtin_amdgcn_global_load_async_to_lds_b128)
    __builtin_amdgcn_global_load_async_to_lds_b128(
        (AS1 v4i*)g, (AS3 v4i*)l, /*imm offset*/0, /*cpol*/0);
#else
    unsigned lds_addr = (unsigned)(size_t)(AS3 void*)l;
    asm volatile("global_load_async_to_lds_b128 %0, %1, off"
                 :: "v"(lds_addr), "v"(g) : "memory");
#endif
}

__device__ __forceinline__ void wait_async_zero() {
#if __has_builtin(__builtin_amdgcn_s_wait_asynccnt)
    __builtin_amdgcn_s_wait_asynccnt(0);
#else
    asm volatile("s_wait_asynccnt 0" ::: "memory");
#endif
}

// ---------------------------------------------------------------------------
// Kernel 1: gating coefficients c[b,e] = sum_f softmax(...)[b,f] * align_conv[f,e]
// One wave per sample row. h0 == 0 in the reference, so W_hh drops out.
// ---------------------------------------------------------------------------
__global__ __launch_bounds__(256) void attn_kernel(
    const float* __restrict__ x, const float* __restrict__ align_conv,
    const float* __restrict__ W_ih, const float* __restrict__ b_ih,
    const float* __restrict__ b_hh, const float* __restrict__ W_att,
    const float* __restrict__ b_att, float* __restrict__ c_out)
{
    const int wave = threadIdx.x >> 5;
    const int lane = threadIdx.x & 31;
    const int b = blockIdx.x * 8 + wave;

    const float* xr = x + (size_t)b * IN_DIM;
    float s = 0.f;
    for (int i = lane; i < IN_DIM; i += 32) s += xr[i];
#pragma unroll
    for (int off = 16; off > 0; off >>= 1) s += __shfl_xor(s, off, 32);

    if (lane == 0) {
        const float pooled = s * (1.0f / (float)IN_DIM);
        float h[E_DIM];
#pragma unroll
        for (int j = 0; j < E_DIM; ++j) {
            float v = tanhf(pooled * W_ih[j] + b_ih[j] + b_hh[j]);
            h[j] = v > 0.f ? v : 0.f;   // ReLU fused
        }
        float logit[E_DIM];
        float mx = -1e30f;
#pragma unroll
        for (int eo = 0; eo < E_DIM; ++eo) {
            float l = b_att[eo];
#pragma unroll
            for (int j = 0; j < E_DIM; ++j) l += h[j] * W_att[eo * E_DIM + j];
            logit[eo] = l;
            mx = fmaxf(mx, l);
        }
        float sum = 0.f;
#pragma unroll
        for (int eo = 0; eo < E_DIM; ++eo) { logit[eo] = expf(logit[eo] - mx); sum += logit[eo]; }
        const float inv = 1.0f / sum;
#pragma unroll
        for (int e2 = 0; e2 < E_DIM; ++e2) {
            float cv = 0.f;
#pragma unroll
            for (int f = 0; f < E_DIM; ++f) cv += (logit[f] * inv) * align_conv[f * E_DIM + e2];
            c_out[b * E_DIM + e2] = cv;
        }
    }
}

// ---------------------------------------------------------------------------
// Kernel 2: per-expert GEMM partials via V_WMMA_F32_16X16X4_F32 with
// double-buffered LDS fed by GLOBAL_LOAD_ASYNC_TO_LDS_B128 DMA.
//   part[e][b][o] = c[b,e] * sum_k x[b,k] * weight[e,o,k]
// Grid: (OUT/64, E). Block: 256 threads = 8 waves.
// Each workgroup owns all 64 B-rows x 64 O-cols of one expert -> weight is
// streamed from HBM exactly once, coalesced 128B segments, DMA'd into LDS.
// Pipeline per K-tile: s_wait_asynccnt 0 -> barrier -> issue DMA for tile i+1
// into the other buffer -> 16 WMMAs on tile i. One barrier per tile.
// ---------------------------------------------------------------------------
__global__ __launch_bounds__(256) void moe_gemm_kernel(
    const float* __restrict__ x, const float* __restrict__ weight,
    const float* __restrict__ c_coef, float* __restrict__ part)
{
    __shared__ float sx[2][B_DIM * LDSW];
    __shared__ float sw[2][B_DIM * LDSW];
    __shared__ float sc[B_DIM];

    const int tid   = threadIdx.x;
    const int e     = blockIdx.y;
    const int o0    = blockIdx.x * 64;
    const int wave  = tid >> 5;
    const int lane  = tid & 31;
    const int nsub  = wave & 3;
    const int mbase = (wave >> 2) * 32;
    const int koff  = (lane >= 16) ? 2 : 0;   // f32 A/B frag K-offset per lane half
    const int lrow  = lane & 15;

    if (tid < B_DIM) sc[tid] = c_coef[tid * E_DIM + e];

    const float* wsrc = weight + (size_t)e * OUT_DIM * IN_DIM;

    // DMA addressing: 512 float4 per tile per array; thread t covers
    // rows (t>>3) and (t>>3)+32 at float-column ((t&7)*4).
    const int ldrow = tid >> 3;
    const int ldcol = (tid & 7) << 2;

    v8f acc0 = {};
    v8f acc1 = {};

    // Prologue: DMA tile 0 into buffer 0.
    {
        const int k0 = 0, buf = 0;
        async_copy_b128(x    + (size_t)ldrow        * IN_DIM + k0 + ldcol, &sx[buf][ ldrow       * LDSW + ldcol]);
        async_copy_b128(x    + (size_t)(ldrow + 32) * IN_DIM + k0 + ldcol, &sx[buf][(ldrow + 32) * LDSW + ldcol]);
        async_copy_b128(wsrc + (size_t)(o0 + ldrow)      * IN_DIM + k0 + ldcol, &sw[buf][ ldrow       * LDSW + ldcol]);
        async_copy_b128(wsrc + (size_t)(o0 + ldrow + 32) * IN_DIM + k0 + ldcol, &sw[buf][(ldrow + 32) * LDSW + ldcol]);
    }

    int cur = 0;
    for (int k0 = 0; k0 < IN_DIM; k0 += KT) {
        wait_async_zero();      // current tile's DMAs (issued last iter) complete
        __syncthreads();        // all waves' DMAs visible; prev compute done

        if (k0 + KT < IN_DIM) { // issue DMA for next tile into the other buffer
            const int kn = k0 + KT, buf = cur ^ 1;
            async_copy_b128(x    + (size_t)ldrow        * IN_DIM + kn + ldcol, &sx[buf][ ldrow       * LDSW + ldcol]);
            async_copy_b128(x    + (size_t)(ldrow + 32) * IN_DIM + kn + ldcol, &sx[buf][(ldrow + 32) * LDSW + ldcol]);
            async_copy_b128(wsrc + (size_t)(o0 + ldrow)      * IN_DIM + kn + ldcol, &sw[buf][ ldrow       * LDSW + ldcol]);
            async_copy_b128(wsrc + (size_t)(o0 + ldrow + 32) * IN_DIM + kn + ldcol, &sw[buf][(ldrow + 32) * LDSW + ldcol]);
        }

        const float* bx = sx[cur];
        const float* bw = sw[cur];
#pragma unroll
        for (int kk = 0; kk < KT; kk += 4) {
            const v2f bfrag  = *(const v2f*)&bw[(nsub * 16 + lrow) * LDSW + kk + koff];
            const v2f afrag0 = *(const v2f*)&bx[(mbase      + lrow) * LDSW + kk + koff];
            const v2f afrag1 = *(const v2f*)&bx[(mbase + 16 + lrow) * LDSW + kk + koff];
            acc0 = __builtin_amdgcn_wmma_f32_16x16x4_f32(
                false, afrag0, false, bfrag, (short)0, acc0, false, false);
            acc1 = __builtin_amdgcn_wmma_f32_16x16x4_f32(
                false, afrag1, false, bfrag, (short)0, acc1, false, false);
        }
        cur ^= 1;
    }

    // Scale by c[b,e] and emit partials. C/D layout: VGPR j -> M = j + 8*(lane>=16).
    const int   hi   = (lane >= 16) ? 8 : 0;
    const int   ncol = o0 + nsub * 16 + lrow;
    float* pdst = part + (size_t)e * B_DIM * OUT_DIM;
#pragma unroll
    for (int j = 0; j < 8; ++j) {
        const int b0 = mbase + j + hi;
        const int b1 = mbase + 16 + j + hi;
        pdst[(size_t)b0 * OUT_DIM + ncol] = acc0[j] * sc[b0];
        pdst[(size_t)b1 * OUT_DIM + ncol] = acc1[j] * sc[b1];
    }
}

// ---------------------------------------------------------------------------
// Kernel 3: y[b,o] = sum_e part[e][b][o]  (deterministic, no atomics)
// ---------------------------------------------------------------------------
__global__ __launch_bounds__(256) void reduce_kernel(
    const float* __restrict__ part, float* __restrict__ out)
{
    const int t = blockIdx.x * 256 + threadIdx.x;   // 65536 outputs
    float s = 0.f;
#pragma unroll
    for (int e = 0; e < E_DIM; ++e) s += part[(size_t)e * (B_DIM * OUT_DIM) + t];
    out[t] = s;
}

// ---------------------------------------------------------------------------
extern "C" void kernel_launch(void* const* d_in, const int* in_sizes, int n_in,
                              void* d_out, int out_size, void* d_ws, size_t ws_size,
                              hipStream_t stream) {
    (void)in_sizes; (void)n_in; (void)out_size; (void)ws_size;
    const float* x          = (const float*)d_in[0];
    const float* weight     = (const float*)d_in[1];
    const float* align_conv = (const float*)d_in[2];
    const float* W_ih       = (const float*)d_in[3];
    // d_in[4] = W_hh: unused (h0 == 0 in the reference forward pass)
    const float* b_ih       = (const float*)d_in[5];
    const float* b_hh       = (const float*)d_in[6];
    const float* W_att      = (const float*)d_in[7];
    const float* b_att      = (const float*)d_in[8];
    float* out = (float*)d_out;

    // Workspace: [E][B][OUT] f32 partials (2 MB) + c[B][E] (512 floats).
    float* part   = (float*)d_ws;
    float* c_coef = part + (size_t)E_DIM * B_DIM * OUT_DIM;

    attn_kernel<<<8, 256, 0, stream>>>(x, align_conv, W_ih, b_ih, b_hh,
                                       W_att, b_att, c_coef);

    dim3 grid(OUT_DIM / 64, E_DIM);
    moe_gemm_kernel<<<grid, 256, 0, stream>>>(x, weight, c_coef, part);

    reduce_kernel<<<(B_DIM * OUT_DIM) / 256, 256, 0, stream>>>(part, out);
}